// _DecorrBatchNorm_44676249813445
// MI455X (gfx1250) — compile-verified
//
#include <hip/hip_runtime.h>

// ---------------------------------------------------------------------------
// DecorrBatchNorm (ZCA whitening) for MI455X / gfx1250.
//  x: [64, 256, 56, 56] fp32 -> 16 groups x 16 features x m=200704 samples.
//  Pass 1: per-group mean + Gram via V_WMMA_F32_16X16X4_F32 (fp32 WMMA).
//  Pass 2: inverse sqrt of 16x16 covariance via coupled Newton-Schulz
//          (mathematically == rot diag(eig^-1/2) rot^T for SPD cov).
//  Pass 3: out = du * x - du*mean, tiled 16x16 through fp32 WMMA.
// Bandwidth-bound: ~617 MB total traffic -> ~26 us floor at 23.3 TB/s.
// ---------------------------------------------------------------------------

typedef __attribute__((ext_vector_type(2))) float v2f;
typedef __attribute__((ext_vector_type(4))) float v4f;
typedef __attribute__((ext_vector_type(8))) float v8f;

#define GN   16
#define FN   16
#define CN   256
#define HWN  3136          // 56*56
#define BN   64
#define MTOT (BN * HWN)    // 200704 samples per (group,feature)
#define EPSV 1e-7f

// workspace layout (floats):
#define WS_P    0          // Gram sums   [16][16][16]
#define WS_S    4096       // feature sums [16][16]
#define WS_DU   4352       // whitening matrices [16][16][16]
#define WS_BIAS 8448       // du*mean       [16][16]
#define WS_ZERO_N 4352     // P + S must be zeroed each call

__device__ __forceinline__ v8f wmma_f32(v2f a, v2f b, v8f c) {
  // V_WMMA_F32_16X16X4_F32 : D(16x16 f32) = A(16x4 f32) * B(4x16 f32) + C
  return __builtin_amdgcn_wmma_f32_16x16x4_f32(
      /*neg_a=*/false, a, /*neg_b=*/false, b,
      /*c_mod=*/(short)0, c, /*reuse_a=*/false, /*reuse_b=*/false);
}

__global__ void zero_ws_kernel(float* __restrict__ ws) {
  int i = blockIdx.x * 256 + threadIdx.x;
  if (i < WS_ZERO_N) ws[i] = 0.0f;
}

// ---------------------------------------------------------------------------
// Pass 1: per-(group,batch) block accumulates Gram (16x16) and feature sums.
// One float4 load per lane feeds TWO 16x16x4 Gram WMMAs (A-layout and
// B-layout are lane-symmetric for this shape, so the same registers act as
// both X and X^T; k-permutation cancels in the Gram).
// ---------------------------------------------------------------------------
__global__ void stats_kernel(const float* __restrict__ x, float* __restrict__ ws) {
  __shared__ float lds_p[8][256];
  __shared__ float lds_s[8][32];

  const int g    = blockIdx.x & 15;
  const int b    = blockIdx.x >> 4;
  const int lane = threadIdx.x & 31;
  const int wave = threadIdx.x >> 5;
  const int f    = lane & 15;
  const int kh   = lane >> 4;          // half-wave selects K-pair

  const float* xf = x + (size_t)(b * CN + g * FN + f) * HWN;

  v8f acc0 = {};
  v8f acc1 = {};
  float ssum = 0.0f;

  const int span    = HWN / 8;         // 392 samples per wave
  const int s_begin = wave * span;
  for (int s = s_begin; s < s_begin + span; s += 8) {
    v4f q = *(const v4f*)(xf + s + 4 * kh);     // 16B aligned
    ssum += q.x + q.y + q.z + q.w;
    v2f a0 = {q.x, q.y};
    v2f a1 = {q.z, q.w};
    acc0 = wmma_f32(a0, a0, acc0);              // Gram over 4 samples
    acc1 = wmma_f32(a1, a1, acc1);              // independent chain
  }
#pragma unroll
  for (int r = 0; r < 8; ++r) acc0[r] += acc1[r];

  // D layout: vgpr r, lane L -> Gram[r + 8*(L>>4)][L&15]
#pragma unroll
  for (int r = 0; r < 8; ++r)
    lds_p[wave][(r + 8 * kh) * 16 + f] = acc0[r];
  lds_s[wave][lane] = ssum;
  __syncthreads();

  const int t = threadIdx.x;
  float v = 0.0f;
#pragma unroll
  for (int w = 0; w < 8; ++w) v += lds_p[w][t];
  atomicAdd(&ws[WS_P + g * 256 + t], v);

  if (t < 16) {
    float s2 = 0.0f;
#pragma unroll
    for (int w = 0; w < 8; ++w) s2 += lds_s[w][t] + lds_s[w][t + 16];
    atomicAdd(&ws[WS_S + g * 16 + t], s2);
  }
}

// ---------------------------------------------------------------------------
// Pass 2: 16 blocks x 256 threads (one thread per matrix element).
// cov = P/m - mean mean^T + eps*I ; du = cov^{-1/2} via Newton-Schulz.
// ---------------------------------------------------------------------------
__global__ void solve_kernel(float* __restrict__ ws) {
  __shared__ float Ysh[16][17], Zsh[16][17], Wsh[16][17];
  __shared__ float meansh[16], diag[16], csh;

  const int g = blockIdx.x;
  const int t = threadIdx.x;
  const int i = t >> 4, j = t & 15;
  const float invm = 1.0f / (float)MTOT;

  if (t < 16) meansh[t] = ws[WS_S + g * 16 + t] * invm;
  __syncthreads();

  float cov = ws[WS_P + g * 256 + t] * invm - meansh[i] * meansh[j]
            + ((i == j) ? EPSV : 0.0f);
  if (i == j) diag[i] = cov;
  __syncthreads();
  if (t == 0) {
    float c = 0.0f;
    for (int d = 0; d < 16; ++d) c += diag[d];
    csh = c;
  }
  __syncthreads();
  const float c = csh;

  Ysh[i][j] = cov / c;
  Zsh[i][j] = (i == j) ? 1.0f : 0.0f;
  __syncthreads();

  for (int it = 0; it < 12; ++it) {
    float tv = 0.0f;
#pragma unroll
    for (int k = 0; k < 16; ++k) tv += Zsh[i][k] * Ysh[k][j];
    Wsh[i][j] = 0.5f * (((i == j) ? 3.0f : 0.0f) - tv);
    __syncthreads();
    float y2 = 0.0f, z2 = 0.0f;
#pragma unroll
    for (int k = 0; k < 16; ++k) {
      y2 += Ysh[i][k] * Wsh[k][j];
      z2 += Wsh[i][k] * Zsh[k][j];
    }
    __syncthreads();
    Ysh[i][j] = y2;
    Zsh[i][j] = z2;
    __syncthreads();
  }

  const float rs = rsqrtf(c);                 // du = Z / sqrt(trace)
  ws[WS_DU + g * 256 + t] = Zsh[i][j] * rs;
  __syncthreads();
  if (t < 16) {
    float bsum = 0.0f;
#pragma unroll
    for (int h = 0; h < 16; ++h) bsum += Zsh[t][h] * rs * meansh[h];
    ws[WS_BIAS + g * 16 + t] = bsum;          // du * mean
  }
}

// ---------------------------------------------------------------------------
// Pass 3: out = du * x - du*mean, per 16x16 tile via 4 fp32 WMMAs.
// du held in A-layout registers per wave; C initialized with -bias.
// ---------------------------------------------------------------------------
__global__ void apply_kernel(const float* __restrict__ x,
                             const float* __restrict__ ws,
                             float* __restrict__ out) {
  const int g    = blockIdx.x & 15;
  const int b    = blockIdx.x >> 4;
  const int lane = threadIdx.x & 31;
  const int wave = threadIdx.x >> 5;
  const int f    = lane & 15;
  const int kh   = lane >> 4;

  const float* du   = ws + WS_DU + g * 256;
  const float* bias = ws + WS_BIAS + g * 16;

  // A-layout: lane L, chunk c, vgpr v holds du[m=L&15][4c + 2*(L>>4) + v]
  v2f a[4];
#pragma unroll
  for (int c = 0; c < 4; ++c)
    a[c] = *(const v2f*)(du + f * 16 + 4 * c + 2 * kh);

  v8f cinit;
#pragma unroll
  for (int r = 0; r < 8; ++r) cinit[r] = -bias[r + 8 * kh];

  const size_t base = (size_t)(b * CN + g * FN) * HWN;
  const float* xb = x + base;
  float*       ob = out + base;

  for (int tile = wave; tile < HWN / 16; tile += 8) {
    const int col = tile * 16 + f;            // this lane's sample column
    v8f acc = cinit;
#pragma unroll
    for (int c = 0; c < 4; ++c) {
      // B-layout: lane L, vgpr v holds x[h = 4c + 2*(L>>4) + v][col]
      v2f bv;
      bv.x = xb[(size_t)(4 * c + 2 * kh + 0) * HWN + col];
      bv.y = xb[(size_t)(4 * c + 2 * kh + 1) * HWN + col];
      acc = wmma_f32(a[c], bv, acc);
    }
    // D layout: vgpr r, lane L -> out[f = r + 8*(L>>4)][col]
#pragma unroll
    for (int r = 0; r < 8; ++r)
      ob[(size_t)(r + 8 * kh) * HWN + col] = acc[r];
  }
}

extern "C" void kernel_launch(void* const* d_in, const int* in_sizes, int n_in,
                              void* d_out, int out_size, void* d_ws, size_t ws_size,
                              hipStream_t stream) {
  const float* x  = (const float*)d_in[0];
  float* out      = (float*)d_out;
  float* ws       = (float*)d_ws;

  zero_ws_kernel<<<(WS_ZERO_N + 255) / 256, 256, 0, stream>>>(ws);
  stats_kernel<<<GN * BN, 256, 0, stream>>>(x, ws);
  solve_kernel<<<GN, 256, 0, stream>>>(ws);
  apply_kernel<<<GN * BN, 256, 0, stream>>>(x, ws, out);
}